// TopLeftPool_54357106098212
// MI455X (gfx1250) — compile-verified
//
#include <hip/hip_runtime.h>
#include <cstdint>

// Top-left corner pooling: p[b,c,i,j] = max_{h>=i, w>=j} x[b,c,h,w]
//  == reverse cummax along H, then reverse cummax along W, per (b,c) plane.
//
// 4096 planes of 128x128 f32. Memory-bound: 256MiB in + 256MiB out
// => ~22us at 23.3 TB/s. One workgroup per plane; plane is staged through
// LDS with gfx1250 async global<->LDS DMA (ASYNCcnt-tracked), both scans run
// in LDS with an XOR chunk swizzle so both the column-strided phase-1 access
// and the row-contiguous b128 phase-2 access are bank-conflict free.

#define TLP_AS1 __attribute__((address_space(1)))
#define TLP_AS3 __attribute__((address_space(3)))

#if defined(__HIP_DEVICE_COMPILE__)
  #if __has_builtin(__builtin_amdgcn_global_load_async_to_lds_b128) && \
      __has_builtin(__builtin_amdgcn_global_store_async_from_lds_b128)
    #define TLP_HAS_ASYNC 1
  #else
    #define TLP_HAS_ASYNC 0
  #endif
#else
  #define TLP_HAS_ASYNC 0
#endif

namespace {

constexpr int H = 128;
constexpr int W = 128;
constexpr int NTHREADS = 128;

// Builtin parameter types per hipcc's diagnostic:
//   arg0: 'int __attribute__((vector_size(16))) __device__ *'  (target AS1)
//   arg1: same vector type in LDS (AS3, 32-bit pointer)
typedef int tlp_v4i __attribute__((vector_size(16)));
typedef TLP_AS1 tlp_v4i* tlp_gvp;   // global-AS pointer to 16B vector
typedef TLP_AS3 tlp_v4i* tlp_lvp;   // LDS-AS pointer to 16B vector

// 16B-chunk XOR swizzle within each LDS row. Keeps LDS at exactly 64KiB
// (no padding) while making phase-1 (column walk) conflict-free and
// phase-2 (row-major b128) hit the b128 minimum of 2 bank cycles.
__device__ __forceinline__ int swz(int row, int chunk_col) {
  return chunk_col ^ (row & 31);
}

// Generic global pointer -> AS1 vector pointer (flat global addrs are
// identity-mapped, so an integer round-trip is a no-op).
__device__ __forceinline__ tlp_gvp as_gvp(const float* p) {
  return (tlp_gvp)(uintptr_t)p;
}
// Generic LDS pointer -> AS3 vector pointer. Per the ISA, the shared
// aperture lives in addr[63:32] and addr[31:0] is the LDS byte offset,
// so truncation to 32 bits is exactly the generic->AS3 addrspacecast.
__device__ __forceinline__ tlp_lvp as_lvp(const float* p) {
  return (tlp_lvp)(uint32_t)(uintptr_t)p;
}

__device__ __forceinline__ void g2l_b128(const float* g, float* l) {
#if TLP_HAS_ASYNC
  __builtin_amdgcn_global_load_async_to_lds_b128(as_gvp(g), as_lvp(l), 0, 0);
#else
  *(float4*)l = *(const float4*)g;
#endif
}

__device__ __forceinline__ void l2g_b128(float* g, const float* l) {
#if TLP_HAS_ASYNC
  __builtin_amdgcn_global_store_async_from_lds_b128(as_gvp(g), as_lvp(l), 0, 0);
#else
  *(float4*)g = *(const float4*)l;
#endif
}

__device__ __forceinline__ void wait_async_zero() {
#if TLP_HAS_ASYNC
  #if __has_builtin(__builtin_amdgcn_s_wait_asynccnt)
    __builtin_amdgcn_s_wait_asynccnt(0);
  #else
    asm volatile("s_wait_asynccnt 0" ::: "memory");
  #endif
#endif
}

} // namespace

__global__ __launch_bounds__(NTHREADS)
void TopLeftPool_54357106098212_kernel(const float* __restrict__ x,
                                       float* __restrict__ out) {
  __shared__ float lds[H * W];  // 64 KiB, XOR-swizzled in 16B chunks
  const int tid = threadIdx.x;                       // 0..127
  const size_t base = (size_t)blockIdx.x * (size_t)(H * W);

  // ---- Stage 0: plane -> LDS via async DMA (coalesced 512B per wave-instr)
  #pragma unroll
  for (int k = 0; k < 32; ++k) {
    const int c = (k << 7) + tid;    // float4-chunk id 0..4095
    const int row = c >> 5;          // 32 chunks per row
    const int kc = c & 31;
    g2l_b128(x + base + ((size_t)c << 2),
             &lds[(row << 7) + (swz(row, kc) << 2)]);
  }
  wait_async_zero();
  __syncthreads();

  // ---- Stage 1: reverse cummax along H; thread t owns column t (b32 RMW)
  {
    const int kc = tid >> 2;
    const int jl = tid & 3;
    float run = -__builtin_huge_valf();
    #pragma unroll 8
    for (int i = H - 1; i >= 0; --i) {
      const int idx = (i << 7) + (swz(i, kc) << 2) + jl;
      run = fmaxf(run, lds[idx]);
      lds[idx] = run;
    }
  }
  __syncthreads();

  // ---- Stage 2: reverse cummax along W; thread t owns row t (b128 RMW)
  {
    const int i = tid;
    float carry = -__builtin_huge_valf();
    #pragma unroll 8
    for (int kc = (W / 4) - 1; kc >= 0; --kc) {
      float4* p = (float4*)&lds[(i << 7) + (swz(i, kc) << 2)];
      float4 v = *p;
      const float w3 = fmaxf(v.w, carry);
      const float w2 = fmaxf(v.z, w3);
      const float w1 = fmaxf(v.y, w2);
      const float w0 = fmaxf(v.x, w1);
      carry = w0;
      v.x = w0; v.y = w1; v.z = w2; v.w = w3;
      *p = v;
    }
  }
  __syncthreads();

  // ---- Stage 3: LDS -> plane via async DMA; S_ENDPGM's implicit wait-idle
  //      covers the outstanding async stores.
  #pragma unroll
  for (int k = 0; k < 32; ++k) {
    const int c = (k << 7) + tid;
    const int row = c >> 5;
    const int kc = c & 31;
    l2g_b128(out + base + ((size_t)c << 2),
             &lds[(row << 7) + (swz(row, kc) << 2)]);
  }
}

extern "C" void kernel_launch(void* const* d_in, const int* in_sizes, int n_in,
                              void* d_out, int out_size, void* d_ws, size_t ws_size,
                              hipStream_t stream) {
  (void)n_in; (void)out_size; (void)d_ws; (void)ws_size;
  const float* x = (const float*)d_in[0];
  float* out = (float*)d_out;
  const int planes = in_sizes[0] / (H * W);   // 16*256 = 4096
  dim3 grid((unsigned)planes), block(NTHREADS);
  TopLeftPool_54357106098212_kernel<<<grid, block, 0, stream>>>(x, out);
}